// NW0_73572789780677
// MI455X (gfx1250) — compile-verified
//
#include <hip/hip_runtime.h>
#include <hip/hip_bf16.h>

// ---------------------------------------------------------------------------
// Gaussian-kernel attention, fused flash-style over the KEY axis.
//   B=8, NK=NQ=4096, IN=3, H=128, DV=64
// Dominant cost: attn@VALUE (17.2 GFLOP) + 134M score dots -> WMMA f32 16x16x4.
// Scores are <= 0 after clamping, so exp() never overflows: streaming softmax
// without running-max rescaling (sum >= strongest term, divide once at end).
// ---------------------------------------------------------------------------

#define BB  8
#define NK  4096
#define NQ  4096
#define DV  64

typedef __attribute__((ext_vector_type(2))) float v2f;
typedef __attribute__((ext_vector_type(8))) float v8f;

__device__ inline v8f wmma4(v2f a, v2f b, v8f c) {
  // D = A(16x4) x B(4x16) + C, all f32
  return __builtin_amdgcn_wmma_f32_16x16x4_f32(false, a, false, b, (short)0, c,
                                               false, false);
}

__device__ inline float swap16(float x) {
  // ds_swizzle SWAPX16: xor=0x10, and=0x1f  ->  offset 0x401F
  return __builtin_bit_cast(
      float, __builtin_amdgcn_ds_swizzle(__builtin_bit_cast(int, x), 0x401F));
}

// ---------------------------------------------------------------------------
// Kernel 1: shared MLP (3 -> 128 -> 128 -> 3) + elementwise scale, emits
// augmented float4 (sx, sy, sz, -0.5*|s|^2) per point.  2.2 GFLOP total: VALU.
// 128 threads handle 32 points; hidden layers staged transposed in LDS.
// ---------------------------------------------------------------------------
__global__ __launch_bounds__(128) void NW0_mlp_aug(
    const float* __restrict__ pts,           // [N,3]
    const float* __restrict__ W1, const float* __restrict__ b1,
    const float* __restrict__ W2, const float* __restrict__ b2,
    const float* __restrict__ W3, const float* __restrict__ b3,
    float4* __restrict__ aug) {
  __shared__ __align__(16) float xs[32][4];
  __shared__ __align__(16) float h1[128][32];   // h1T[j][p]
  __shared__ __align__(16) float h2[128][32];   // h2T[u][p]
  __shared__ __align__(16) float sc[32][4];

  const int t    = threadIdx.x;
  const int base = blockIdx.x * 32;

  if (t < 32) {
    const float* g = pts + (size_t)(base + t) * 3;
    xs[t][0] = g[0]; xs[t][1] = g[1]; xs[t][2] = g[2]; xs[t][3] = 0.f;
  }
  __syncthreads();

  // layer 1: thread t == hidden unit u
  {
    const float w0 = W1[t * 3 + 0], w1 = W1[t * 3 + 1], w2 = W1[t * 3 + 2];
    const float bb = b1[t];
    #pragma unroll
    for (int p = 0; p < 32; ++p) {
      float h = fmaf(w0, xs[p][0], fmaf(w1, xs[p][1], fmaf(w2, xs[p][2], bb)));
      h1[t][p] = fmaxf(h, 0.f);
    }
  }
  __syncthreads();

  // layer 2: 128x128, 32 accumulators per thread, LDS rows read as float4
  {
    const float bb = b2[t];
    float4 acc[8];
    #pragma unroll
    for (int g = 0; g < 8; ++g) acc[g] = make_float4(bb, bb, bb, bb);
    const float* w2r = W2 + t * 128;
    for (int j = 0; j < 128; ++j) {
      const float  w   = w2r[j];
      const float4* row = (const float4*)(&h1[j][0]);
      #pragma unroll
      for (int g = 0; g < 8; ++g) {
        float4 h = row[g];
        acc[g].x = fmaf(w, h.x, acc[g].x);
        acc[g].y = fmaf(w, h.y, acc[g].y);
        acc[g].z = fmaf(w, h.z, acc[g].z);
        acc[g].w = fmaf(w, h.w, acc[g].w);
      }
    }
    #pragma unroll
    for (int g = 0; g < 8; ++g) {
      h2[t][4 * g + 0] = fmaxf(acc[g].x, 0.f);
      h2[t][4 * g + 1] = fmaxf(acc[g].y, 0.f);
      h2[t][4 * g + 2] = fmaxf(acc[g].z, 0.f);
      h2[t][4 * g + 3] = fmaxf(acc[g].w, 0.f);
    }
  }
  __syncthreads();

  // layer 3: thread -> (point p = t>>2, channel c = t&3), c<3 active
  {
    const int p = t >> 2, c = t & 3;
    if (c < 3) {
      float acc = b3[c];
      const float* w3r = W3 + c * 128;
      for (int u = 0; u < 128; ++u) acc = fmaf(w3r[u], h2[u][p], acc);
      sc[p][c] = xs[p][c] * acc;   // x * mlp(x)
    }
  }
  __syncthreads();

  if (t < 32) {
    const float sx = sc[t][0], sy = sc[t][1], sz = sc[t][2];
    const float w  = -0.5f * (sx * sx + sy * sy + sz * sz);
    aug[base + t]  = make_float4(sx, sy, sz, w);
  }
}

// ---------------------------------------------------------------------------
// Kernel 2: fused score + softmax + context.
// Block = 256 thr (8 waves); wave w owns 16 queries; loop over 256 key tiles.
// Per key tile per wave: 1 score WMMA + 16 accumulate WMMAs (Vt in LDS),
// 8 exps, 8 ds_swizzle SWAPX16 to re-layout P as a WMMA B operand.
// ---------------------------------------------------------------------------
__global__ __launch_bounds__(256) void NW0_attn(
    const float4* __restrict__ kaug, const float4* __restrict__ qaug,
    const float* __restrict__ value, float* __restrict__ out) {
  __shared__ __align__(16) float vts[16 * 80];   // V tile 16x64, stride 80

  const int b    = blockIdx.x >> 5;       // 32 query blocks per batch
  const int qblk = blockIdx.x & 31;
  const int tid  = threadIdx.x;
  const int wave = tid >> 5;
  const int lane = tid & 31;
  const int l16  = lane & 15;
  const bool hi  = lane >= 16;
  const int q    = qblk * 128 + wave * 16 + l16;  // this lane's query column

  // B operand of the score WMMA: rows (Qx,Qy,Qz,1); C = -0.5*q2 per column.
  const float4 qa = qaug[(size_t)b * NQ + q];
  v2f Bq; Bq.x = hi ? qa.z : qa.x; Bq.y = hi ? 1.0f : qa.y;
  const float cq = qa.w;                       // -0.5*|Qs|^2
  const v8f c8 = {cq, cq, cq, cq, cq, cq, cq, cq};

  v8f o[4] = {v8f{}, v8f{}, v8f{}, v8f{}};     // O^T[v,q], 4 v-tiles of 16
  float lacc = 0.f;                            // partial softmax denominator

  const float*  vb = value + (size_t)b * NK * DV;
  const float4* kb = kaug + (size_t)b * NK;

  for (int kt = 0; kt < NK; kt += 16) {
    // ---- stage V[kt..kt+15][0..63] into LDS (256 thr x float4) ----
    {
      const int r = tid >> 4, c = (tid & 15) << 2;
      const float4 vv = *(const float4*)(vb + (size_t)(kt + r) * DV + c);
      *(float4*)(vts + r * 80 + c) = vv;
    }
    __syncthreads();

    // ---- score tile: D[k,q] = K.Q - 0.5k2 - 0.5q2  (one WMMA) ----
    const float4 ka = kb[kt + l16];            // (Kx,Ky,Kz,-0.5*k2)
    v2f Ak; Ak.x = hi ? ka.z : ka.x; Ak.y = hi ? ka.w : ka.y;
    const v8f s = wmma4(Ak, Bq, c8);

    // ---- clamp (sqdist >= 0) + exp; scores <= 0 so exp in [0,1] ----
    float p0 = __expf(fminf(s[0], 0.f)), p1 = __expf(fminf(s[1], 0.f));
    float p2 = __expf(fminf(s[2], 0.f)), p3 = __expf(fminf(s[3], 0.f));
    float p4 = __expf(fminf(s[4], 0.f)), p5 = __expf(fminf(s[5], 0.f));
    float p6 = __expf(fminf(s[6], 0.f)), p7 = __expf(fminf(s[7], 0.f));
    lacc += ((p0 + p1) + (p2 + p3)) + ((p4 + p5) + (p6 + p7));

    // ---- re-layout P (C-layout) into four 4x16 B operands via SWAPX16 ----
    const float q0 = swap16(p0), q1 = swap16(p1), q2 = swap16(p2),
                q3 = swap16(p3), q4 = swap16(p4), q5 = swap16(p5),
                q6 = swap16(p6), q7 = swap16(p7);
    v2f Bp0; Bp0.x = hi ? q2 : p0; Bp0.y = hi ? q3 : p1;  // k rows 0..3
    v2f Bp1; Bp1.x = hi ? q6 : p4; Bp1.y = hi ? q7 : p5;  // k rows 4..7
    v2f Bp2; Bp2.x = hi ? p2 : q0; Bp2.y = hi ? p3 : q1;  // k rows 8..11
    v2f Bp3; Bp3.x = hi ? p6 : q4; Bp3.y = hi ? p7 : q5;  // k rows 12..15

    // ---- O^T += V^T x P : 4 v-tiles x 4 chained K=4 WMMAs ----
    const int d = hi ? 2 : 0;
    #pragma unroll
    for (int vt = 0; vt < 4; ++vt) {
      const int cb = vt * 16 + l16;
      v2f a0, a1, a2, a3;
      a0.x = vts[(0 + d) * 80 + cb];  a0.y = vts[(1 + d) * 80 + cb];
      a1.x = vts[(4 + d) * 80 + cb];  a1.y = vts[(5 + d) * 80 + cb];
      a2.x = vts[(8 + d) * 80 + cb];  a2.y = vts[(9 + d) * 80 + cb];
      a3.x = vts[(12 + d) * 80 + cb]; a3.y = vts[(13 + d) * 80 + cb];
      v8f acc = o[vt];
      acc = wmma4(a0, Bp0, acc);
      acc = wmma4(a1, Bp1, acc);
      acc = wmma4(a2, Bp2, acc);
      acc = wmma4(a3, Bp3, acc);
      o[vt] = acc;
    }
    __syncthreads();
  }

  // ---- finalize: combine lane halves of the denominator, divide, store ----
  const float inv = 1.0f / (lacc + swap16(lacc));
  float* ob = out + ((size_t)(b * NQ + q)) * DV + (hi ? 8 : 0);
  #pragma unroll
  for (int vt = 0; vt < 4; ++vt) {
    const v8f ov = o[vt] * inv;
    *(float4*)(ob + vt * 16)     = make_float4(ov[0], ov[1], ov[2], ov[3]);
    *(float4*)(ob + vt * 16 + 4) = make_float4(ov[4], ov[5], ov[6], ov[7]);
  }
}

// ---------------------------------------------------------------------------
extern "C" void kernel_launch(void* const* d_in, const int* in_sizes, int n_in,
                              void* d_out, int out_size, void* d_ws,
                              size_t ws_size, hipStream_t stream) {
  const float* KEY   = (const float*)d_in[0];
  const float* VALUE = (const float*)d_in[1];
  const float* QUERY = (const float*)d_in[2];
  const float* W1 = (const float*)d_in[3];
  const float* b1 = (const float*)d_in[4];
  const float* W2 = (const float*)d_in[5];
  const float* b2 = (const float*)d_in[6];
  const float* W3 = (const float*)d_in[7];
  const float* b3 = (const float*)d_in[8];

  float4* kaug = (float4*)d_ws;                 // 512 KB
  float4* qaug = kaug + (size_t)BB * NK;        // 512 KB

  NW0_mlp_aug<<<(BB * NK) / 32, 128, 0, stream>>>(KEY, W1, b1, W2, b2, W3, b3,
                                                  kaug);
  NW0_mlp_aug<<<(BB * NQ) / 32, 128, 0, stream>>>(QUERY, W1, b1, W2, b2, W3,
                                                  b3, qaug);
  NW0_attn<<<BB * (NQ / 128), 256, 0, stream>>>(kaug, qaug, VALUE,
                                                (float*)d_out);
}